// CausalEncoder_68358699483893
// MI455X (gfx1250) — compile-verified
//
#include <hip/hip_runtime.h>

// ---------------------------------------------------------------------------
// CausalEncoder fused kernel for gfx1250 (MI455X).
//
// Algebraic folding: send/recv aggregation + first linear layer collapse into
//   W1eff = A^T @ W1[:10] + A @ W1[10:]     (10x10)
// so the whole network is 3 chained 10->10 GEMMs per row + ReLU, executed as
// V_WMMA_F32_16X16X4_F32 on 16-row tiles (K padded 10->12, N padded 10->16,
// pads are exact zeros). Memory-bound: 320 MB total traffic @ 23.3 TB/s.
// ---------------------------------------------------------------------------

typedef __attribute__((ext_vector_type(2))) float v2f;
typedef __attribute__((ext_vector_type(8))) float v8f;

// d_ws float layout: 4 padded weight matrices [12][16] + 3 padded biases [16]
#define WS_W1   0
#define WS_W2   192
#define WS_W3A  384
#define WS_W3B  576
#define WS_B1   768
#define WS_B2   784
#define WS_B3   800

// -------------------------- prologue: fold weights -------------------------
__global__ void prep_weights(const float* __restrict__ A,
                             const float* __restrict__ W1, const float* __restrict__ b1,
                             const float* __restrict__ W2, const float* __restrict__ b2,
                             const float* __restrict__ W3, const float* __restrict__ b3,
                             float* __restrict__ ws) {
  const int t = threadIdx.x;       // single block of 256
  if (t < 192) {
    const int k = t >> 4;          // 0..11  (K index, padded)
    const int n = t & 15;          // 0..15  (N index, padded)
    float w1v = 0.f, w2v = 0.f, w3av = 0.f, w3bv = 0.f;
    if (k < 10 && n < 10) {
      // W1eff[k][n] = sum_j A[j][k]*W1[j][n] + sum_j A[k][j]*W1[10+j][n]
      float s = 0.f;
      for (int j = 0; j < 10; ++j) {
        s += A[j * 10 + k] * W1[j * 10 + n];          // send path (X @ A^T)
        s += A[k * 10 + j] * W1[(10 + j) * 10 + n];   // recv path (X @ A)
      }
      w1v  = s;
      w2v  = W2[k * 10 + n];
      w3av = W3[k * 10 + n];          // acts on X
      w3bv = W3[(10 + k) * 10 + n];   // acts on He
    }
    ws[WS_W1  + t] = w1v;
    ws[WS_W2  + t] = w2v;
    ws[WS_W3A + t] = w3av;
    ws[WS_W3B + t] = w3bv;
  }
  if (t < 16) {
    ws[WS_B1 + t] = (t < 10) ? b1[t] : 0.f;
    ws[WS_B2 + t] = (t < 10) ? b2[t] : 0.f;
    ws[WS_B3 + t] = (t < 10) ? b3[t] : 0.f;
  }
}

// ------------------------------ main kernel --------------------------------
__device__ __forceinline__ v8f wmma4(v2f a, v2f b, v8f c) {
  // D(16x16,f32) = A(16x4,f32) x B(4x16,f32) + C
  return __builtin_amdgcn_wmma_f32_16x16x4_f32(false, a, false, b, 0, c, false, false);
}

__device__ __forceinline__ v8f relu8(v8f x) {
#pragma unroll
  for (int i = 0; i < 8; ++i) x[i] = fmaxf(x[i], 0.f);
  return x;
}

// RAW handoff through LDS (store -> cross-lane load): DS pipe is in-order per
// wave; hard wait + compiler fence keeps everything honest.
#define LDS_FENCE() asm volatile("s_wait_dscnt 0" ::: "memory")
// WAR handoff (load -> overwriting store): hardware DS in-order already makes
// this safe; only compiler reordering must be prevented. No wave stall.
#define LDS_SOFT_FENCE() asm volatile("" ::: "memory")

#define SX_STRIDE 12   // 16x12 X tile, K=10,11 columns pre-zeroed; conflict-free
#define SH_STRIDE 17   // 16x16 staging with stride 17: kills lane/lane+16 aliasing
#define WAVE_LDS  (16 * SX_STRIDE + 16 * SH_STRIDE)

__global__ void __launch_bounds__(256, 2)
causal_encoder_kernel(const float* __restrict__ X, const float* __restrict__ ws,
                      float* __restrict__ Z, int ntiles, long long nelem) {
  __shared__ float lds[8][WAVE_LDS];
  const int lane = threadIdx.x & 31;
  const int wave = threadIdx.x >> 5;
  const int half = lane >> 4;   // 0: lanes 0-15, 1: lanes 16-31
  const int l16  = lane & 15;
  float* sx = lds[wave];                       // [16][SX_STRIDE] X tile
  float* sh = lds[wave] + 16 * SX_STRIDE;      // [16][SH_STRIDE] h/He/Z staging

  // Zero the K=10,11 pad columns once; they are never written again.
  if (lane < 16) {
    sx[lane * SX_STRIDE + 10] = 0.f;
    sx[lane * SX_STRIDE + 11] = 0.f;
  }

  // ---- load weights in WMMA B-layout: chunk c vgpr j lane -> W[4c+j+2*half][l16]
  v2f w1[3], w2[3], w3a[3], w3b[3];
#pragma unroll
  for (int c = 0; c < 3; ++c) {
    const int r = 4 * c + 2 * half;
    w1[c]  = v2f{ws[WS_W1  + r * 16 + l16], ws[WS_W1  + (r + 1) * 16 + l16]};
    w2[c]  = v2f{ws[WS_W2  + r * 16 + l16], ws[WS_W2  + (r + 1) * 16 + l16]};
    w3a[c] = v2f{ws[WS_W3A + r * 16 + l16], ws[WS_W3A + (r + 1) * 16 + l16]};
    w3b[c] = v2f{ws[WS_W3B + r * 16 + l16], ws[WS_W3B + (r + 1) * 16 + l16]};
  }
  const float b1v = ws[WS_B1 + l16];
  const float b2v = ws[WS_B2 + l16];
  const float b3v = ws[WS_B3 + l16];

  // Loop-invariant staging offsets (lane-only; hoisted once):
  //   X stage write:  flat idx -> padded [row][k] slot
  //   Z stage read:   padded [row][n] slot -> flat idx
  int sxoff[5], zoff[5];
#pragma unroll
  for (int i = 0; i < 5; ++i) {
    const int idx = lane + 32 * i;                    // 0..159 within tile
    sxoff[i] = (idx / 10) * SX_STRIDE + (idx % 10);
    zoff[i]  = (idx / 10) * SH_STRIDE + (idx % 10);
  }
  LDS_FENCE();   // pad-zero stores visible before first A-layout reads

  const int gwave  = blockIdx.x * 8 + wave;
  const int nwaves = gridDim.x * 8;

  for (int t = gwave; t < ntiles; t += nwaves) {
    const long long base = (long long)t * 160;
    const bool full = (base + 160) <= nelem;          // wave-uniform

    // ---- coalesced stage of the 16x10 X tile into LDS (5 b32 per lane)
    if (full) {
#pragma unroll
      for (int i = 0; i < 5; ++i) sx[sxoff[i]] = X[base + lane + 32 * i];
    } else {
#pragma unroll
      for (int i = 0; i < 5; ++i) {
        const long long g = base + lane + 32 * i;
        sx[sxoff[i]] = (g < nelem) ? X[g] : 0.f;
      }
    }
    // prefetch next tile for this wave (speculative, dropped if OOB)
    {
      long long nb = base + (long long)nwaves * 160 + lane * 5;
      if (nb < nelem) __builtin_prefetch(&X[nb], 0, 0);
    }
    LDS_FENCE();

    // ---- X in WMMA A-layout (unconditional: pad columns are zero in LDS)
    v2f xa[3];
    {
      const int rb = l16 * SX_STRIDE + 2 * half;
      xa[0] = v2f{sx[rb],     sx[rb + 1]};
      xa[1] = v2f{sx[rb + 4], sx[rb + 5]};
      xa[2] = v2f{sx[rb + 8], sx[rb + 9]};   // half=1 reads K=10,11 -> zeros
    }

    // ---- GEMM1: h = relu(X @ W1eff + b1)
    v8f h = {b1v, b1v, b1v, b1v, b1v, b1v, b1v, b1v};
    h = wmma4(xa[0], w1[0], h);
    h = wmma4(xa[1], w1[1], h);
    h = wmma4(xa[2], w1[2], h);
    h = relu8(h);
#pragma unroll
    for (int v = 0; v < 8; ++v) sh[(v + 8 * half) * SH_STRIDE + l16] = h[v];
    LDS_FENCE();

    // ---- h back in A-layout
    v2f ha[3];
#pragma unroll
    for (int c = 0; c < 3; ++c) {
      const int a = l16 * SH_STRIDE + 4 * c + 2 * half;
      ha[c] = v2f{sh[a], sh[a + 1]};
    }
    LDS_SOFT_FENCE();

    // ---- GEMM2: He = relu(h @ W2 + b2)
    v8f he = {b2v, b2v, b2v, b2v, b2v, b2v, b2v, b2v};
    he = wmma4(ha[0], w2[0], he);
    he = wmma4(ha[1], w2[1], he);
    he = wmma4(ha[2], w2[2], he);
    he = relu8(he);
#pragma unroll
    for (int v = 0; v < 8; ++v) sh[(v + 8 * half) * SH_STRIDE + l16] = he[v];
    LDS_FENCE();

    v2f hea[3];
#pragma unroll
    for (int c = 0; c < 3; ++c) {
      const int a = l16 * SH_STRIDE + 4 * c + 2 * half;
      hea[c] = v2f{sh[a], sh[a + 1]};
    }
    LDS_SOFT_FENCE();

    // ---- GEMM3: Z = relu(X @ W3a + He @ W3b + b3), fused via accumulator chain
    v8f z = {b3v, b3v, b3v, b3v, b3v, b3v, b3v, b3v};
    z = wmma4(xa[0],  w3a[0], z);
    z = wmma4(xa[1],  w3a[1], z);
    z = wmma4(xa[2],  w3a[2], z);
    z = wmma4(hea[0], w3b[0], z);
    z = wmma4(hea[1], w3b[1], z);
    z = wmma4(hea[2], w3b[2], z);
    z = relu8(z);
#pragma unroll
    for (int v = 0; v < 8; ++v) sh[(v + 8 * half) * SH_STRIDE + l16] = z[v];
    LDS_FENCE();

    // ---- coalesced store of the 16x10 Z tile (5 b32 per lane)
    if (full) {
#pragma unroll
      for (int i = 0; i < 5; ++i) Z[base + lane + 32 * i] = sh[zoff[i]];
    } else {
#pragma unroll
      for (int i = 0; i < 5; ++i) {
        const long long g = base + lane + 32 * i;
        if (g < nelem) Z[g] = sh[zoff[i]];
      }
    }
    LDS_SOFT_FENCE();   // keep next iteration's stage stores behind these reads
  }
}

// ------------------------------- launcher ----------------------------------
extern "C" void kernel_launch(void* const* d_in, const int* in_sizes, int n_in,
                              void* d_out, int out_size, void* d_ws, size_t ws_size,
                              hipStream_t stream) {
  const float* X  = (const float*)d_in[0];
  const float* A  = (const float*)d_in[1];
  const float* W1 = (const float*)d_in[2];
  const float* b1 = (const float*)d_in[3];
  const float* W2 = (const float*)d_in[4];
  const float* b2 = (const float*)d_in[5];
  const float* W3 = (const float*)d_in[6];
  const float* b3 = (const float*)d_in[7];
  float* ws = (float*)d_ws;
  float* Z  = (float*)d_out;

  const long long nelem = (long long)in_sizes[0];          // B * 10
  const int ntiles = (int)((nelem + 159) / 160);           // 16-row tiles

  prep_weights<<<1, 256, 0, stream>>>(A, W1, b1, W2, b2, W3, b3, ws);

  int grid = (ntiles + 7) / 8;       // 8 waves per block
  if (grid > 2048) grid = 2048;      // grid-stride over tiles
  if (grid < 1) grid = 1;
  causal_encoder_kernel<<<grid, 256, 0, stream>>>(X, ws, Z, ntiles, nelem);
}